// HNOSegXS_22368189677788
// MI455X (gfx1250) — compile-verified
//
#include <hip/hip_runtime.h>
#include <hip/hip_bf16.h>

typedef __attribute__((ext_vector_type(2))) float v2f;
typedef __attribute__((ext_vector_type(8))) float v8f;

#define WAVES 8   // 8 wave32 per 256-thread block

__device__ __forceinline__ v8f wmma_f32_4(v2f a, v2f b, v8f c) {
  // V_WMMA_F32_16X16X4_F32: D = A(16x4) * B(4x16) + C(16x16), fp32 throughout.
  return __builtin_amdgcn_wmma_f32_16x16x4_f32(false, a, false, b, (short)0, c,
                                               false, false);
}

// ---------------------------------------------------------------------------
// Hartley matrices. Hf[24][96]: crop-fused forward cas matrix rows are the
// kept corner frequencies {0..11, 84..95}. Hi[96][24]: pad-fused inverse,
// columns are corner positions, scaled by 1/96 (one axis worth of 1/N^3).
// ---------------------------------------------------------------------------
__global__ void init_hartley_k(float* __restrict__ Hf, float* __restrict__ Hi) {
  int t = blockIdx.x * blockDim.x + threadIdx.x;
  const float w = 6.283185307179586f / 96.0f;
  if (t < 24 * 96) {
    int k = t / 96, n = t % 96;
    int freq = (k < 12) ? k : (k + 72);
    int m = (freq * n) % 96;                 // exact range reduction
    float a = w * (float)m;
    Hf[t] = cosf(a) + sinf(a);
  } else if (t < 24 * 96 + 96 * 24) {
    int t2 = t - 24 * 96;
    int n = t2 / 24, k = t2 % 24;
    int freq = (k < 12) ? k : (k + 72);
    int m = (freq * n) % 96;
    float a = w * (float)m;
    Hi[t2] = (cosf(a) + sinf(a)) * (1.0f / 96.0f);
  }
}

// ---------------------------------------------------------------------------
// Generic 1D transform along one tensor axis via WMMA.
//   out[line, q] = sum_n in[line, n] * Mtx[q*Kin + n]
// Lines decompose as (hi, lo): in offset = hi*Kin*LoSize + n*LoSize + lo,
// out offset = hi*Kout*LoSize + q*LoSize + lo. Mtx (Kout x Kin) staged in LDS.
// Grid: x = numLines/16/WAVES (exact), y = ceil(Kout/16) with tail masking.
// ---------------------------------------------------------------------------
__global__ void dht_axis_wmma(const float* __restrict__ in, float* __restrict__ out,
                              const float* __restrict__ Mtx,
                              int Kin, int Kout, int LoSize) {
  __shared__ float sB[2304];                 // Kin*Kout == 96*24 both directions
  int total = Kin * Kout;
  for (int i = threadIdx.x; i < total; i += blockDim.x) sB[i] = Mtx[i];
  __syncthreads();

  int lane = threadIdx.x & 31;
  int wv   = threadIdx.x >> 5;
  int hh   = lane >> 4;                      // which 16-lane half
  int ln   = lane & 15;

  long mtile = (long)blockIdx.x * WAVES + wv;
  long m  = mtile * 16 + ln;                 // this lane's A-row (M index)
  long hi = m / LoSize;
  int  lo = (int)(m - hi * (long)LoSize);
  const float* inp = in + hi * (long)Kin * LoSize + lo;

  int q  = blockIdx.y * 16 + ln;             // output frequency (N index)
  int qc = q < Kout ? q : Kout - 1;          // clamp for tail tile (stores masked)
  const float* brow = &sB[qc * Kin];

  v8f acc = {};
  for (int r = 0; r < Kin; r += 4) {
    int k0 = r + hh * 2;                     // A/B K per ISA lane layout
    v2f a, b;
    a.x = inp[(long)k0 * LoSize];
    a.y = inp[(long)(k0 + 1) * LoSize];
    b.x = brow[k0];
    b.y = brow[k0 + 1];
    acc = wmma_f32_4(a, b, acc);
  }

  if (q < Kout) {
    long outq = (long)q * LoSize;
    #pragma unroll
    for (int rr = 0; rr < 8; ++rr) {         // D VGPR rr -> M = rr + 8*hh
      long mm = mtile * 16 + rr + hh * 8;
      long h2 = mm / LoSize;
      int  l2 = (int)(mm - h2 * (long)LoSize);
      out[h2 * (long)Kout * LoSize + outq + l2] = acc[rr];
    }
  }
}

// ---------------------------------------------------------------------------
// 1x1x1 conv as GEMM: out[o, s] = sum_i X[i, s] * W[i*Cout+o] + bias[o].
// Channels >= Csplit come from X1 (for the concat conv). W staged in LDS.
// Grid: x = S/16/WAVES (exact), y = Cout/16.
// ---------------------------------------------------------------------------
__global__ void conv1x1_wmma(const float* __restrict__ X0, const float* __restrict__ X1,
                             const float* __restrict__ W, const float* __restrict__ bias,
                             float* __restrict__ out, long S, int Cin, int Csplit,
                             int Cout) {
  __shared__ float sW[8192];                 // up to 128x64 weights
  int total = Cin * Cout;
  for (int i = threadIdx.x; i < total; i += blockDim.x) sW[i] = W[i];
  __syncthreads();

  int lane = threadIdx.x & 31;
  int wv   = threadIdx.x >> 5;
  int hh   = lane >> 4;
  int ln   = lane & 15;

  long s0 = ((long)blockIdx.x * WAVES + wv) * 16;
  int  q  = blockIdx.y * 16 + ln;            // output channel

  float bv = bias[q];
  v8f acc;
  #pragma unroll
  for (int i = 0; i < 8; ++i) acc[i] = bv;   // bias folded into C

  long sA = s0 + ln;                         // this lane's spatial column (M)
  for (int r = 0; r < Cin; r += 4) {
    int k0 = r + hh * 2;
    const float* p0 = (k0 < Csplit)     ? (X0 + (long)k0 * S)
                                        : (X1 + (long)(k0 - Csplit) * S);
    const float* p1 = (k0 + 1 < Csplit) ? (X0 + (long)(k0 + 1) * S)
                                        : (X1 + (long)(k0 + 1 - Csplit) * S);
    v2f a, b;
    a.x = p0[sA];
    a.y = p1[sA];
    b.x = sW[k0 * Cout + q];
    b.y = sW[(k0 + 1) * Cout + q];
    acc = wmma_f32_4(a, b, acc);
  }

  long base = (long)q * S + s0 + hh * 8;     // 8 contiguous spatial floats
  #pragma unroll
  for (int rr = 0; rr < 8; ++rr) out[base + rr] = acc[rr];
}

// ---------------------------------------------------------------------------
// Spectral block: out[o,p] = xc[o,p] + sum_i xc[i,p]*sw[(i*C+o)*P+p]
//                            + sum_i xc[i,p]*cw[i*C+o]
// Pure streaming over the 226MB sw tensor -> HBM bound; prefetch ahead.
// Grid: x = P/256, y = C.
// ---------------------------------------------------------------------------
__global__ void spectral_block_k(const float* __restrict__ xc, const float* __restrict__ sw,
                                 const float* __restrict__ cw, float* __restrict__ out,
                                 int C, long P) {
  long p = (long)blockIdx.x * blockDim.x + threadIdx.x;
  int  o = blockIdx.y;
  float acc = xc[(long)o * P + p];           // residual
  const float* swp = sw + (long)o * P + p;
  for (int i = 0; i < C; ++i) {
    float xv = xc[(long)i * P + p];
    float wv = swp[(long)i * C * P];
    __builtin_prefetch(swp + (long)(i + 2) * C * P, 0, 1);
    acc = fmaf(xv, wv, acc);
    acc = fmaf(xv, cw[i * C + o], acc);
  }
  out[(long)o * P + p] = acc;
}

// ---------------------------------------------------------------------------
// GroupNorm(1 group, B=1): global mean/var + affine + ReLU (all GNs here relu).
// ---------------------------------------------------------------------------
__global__ void reset_stats_k(float* stats) { stats[0] = 0.f; stats[1] = 0.f; }

__global__ void gn_stats_k(const float* __restrict__ x, long n,
                           float* __restrict__ stats) {
  __shared__ float ss[256], sq[256];
  float s = 0.f, s2 = 0.f;
  for (long i = (long)blockIdx.x * blockDim.x + threadIdx.x; i < n;
       i += (long)gridDim.x * blockDim.x) {
    float v = x[i];
    s += v;
    s2 = fmaf(v, v, s2);
  }
  int tid = threadIdx.x;
  ss[tid] = s; sq[tid] = s2;
  __syncthreads();
  for (int o = 128; o > 0; o >>= 1) {
    if (tid < o) { ss[tid] += ss[tid + o]; sq[tid] += sq[tid + o]; }
    __syncthreads();
  }
  if (tid == 0) { atomicAdd(&stats[0], ss[0]); atomicAdd(&stats[1], sq[0]); }
}

__global__ void gn_apply_k(float* __restrict__ x, long S, long n,
                           const float* __restrict__ w, const float* __restrict__ b,
                           const float* __restrict__ stats) {
  float inv_n = 1.0f / (float)n;
  float mean  = stats[0] * inv_n;
  float var   = stats[1] * inv_n - mean * mean;
  float rstd  = rsqrtf(var + 1e-5f);
  for (long i = (long)blockIdx.x * blockDim.x + threadIdx.x; i < n;
       i += (long)gridDim.x * blockDim.x) {
    long c = i / S;
    float v = (x[i] - mean) * rstd * w[c] + b[c];
    x[i] = fmaxf(v, 0.f);
  }
}

// ---------------------------------------------------------------------------
extern "C" void kernel_launch(void* const* d_in, const int* in_sizes, int n_in,
                              void* d_out, int out_size, void* d_ws, size_t ws_size,
                              hipStream_t stream) {
  const float* x     = (const float*)d_in[0];
  const float* Wm    = (const float*)d_in[1];
  const float* bm    = (const float*)d_in[2];
  const float* gnm_w = (const float*)d_in[3];
  const float* gnm_b = (const float*)d_in[4];
  const float* sw0   = (const float*)d_in[5];
  const float* cw0   = (const float*)d_in[6];
  const float* gn0_w = (const float*)d_in[7];
  const float* gn0_b = (const float*)d_in[8];
  const float* sw1   = (const float*)d_in[9];
  const float* cw1   = (const float*)d_in[10];
  const float* gn1_w = (const float*)d_in[11];
  const float* gn1_b = (const float*)d_in[12];
  const float* gno_w = (const float*)d_in[13];
  const float* gno_b = (const float*)d_in[14];
  const float* Wc    = (const float*)d_in[15];
  const float* bc    = (const float*)d_in[16];
  const float* gnc_w = (const float*)d_in[17];
  const float* gnc_b = (const float*)d_in[18];
  float* out = (float*)d_out;
  (void)in_sizes; (void)n_in; (void)out_size; (void)ws_size;

  const long S = 96L * 96 * 96;   // 884736
  const long P = 24L * 24 * 24;   // 13824
  const int  Cin = 32, C = 64;

  char* base = (char*)d_ws;
  size_t off = 0;
  auto walloc = [&](size_t bytes) -> void* {
    void* p = base + off;
    off = (off + bytes + 255) & ~(size_t)255;
    return p;
  };
  float* Hf    = (float*)walloc(24 * 96 * sizeof(float));
  float* Hi    = (float*)walloc(96 * 24 * sizeof(float));
  float* stats = (float*)walloc(2 * sizeof(float));
  float* bufA  = (float*)walloc((size_t)C * S * sizeof(float));       // mapped "tmp"
  float* bufB  = (float*)walloc((size_t)C * S * sizeof(float));       // inverse result
  float* t1    = (float*)walloc((size_t)C * 96 * 96 * 24 * sizeof(float));
  float* t2    = (float*)walloc((size_t)C * 96 * 24 * 24 * sizeof(float));
  float* xcA   = (float*)walloc((size_t)C * P * sizeof(float));
  float* xcB   = (float*)walloc((size_t)C * P * sizeof(float));
  float* u1    = (float*)walloc((size_t)C * 96 * 24 * 24 * sizeof(float));
  float* u2    = (float*)walloc((size_t)C * 96 * 96 * 24 * sizeof(float));

  dim3 blk(256);

  init_hartley_k<<<dim3(18), blk, 0, stream>>>(Hf, Hi);

  // 1) mapping conv 32->64 + bias, then GN + relu (in place)
  conv1x1_wmma<<<dim3(6912, 4), blk, 0, stream>>>(x, nullptr, Wm, bm, bufA, S, Cin, Cin, C);
  reset_stats_k<<<1, 1, 0, stream>>>(stats);
  gn_stats_k<<<dim3(2048), blk, 0, stream>>>(bufA, C * S, stats);
  gn_apply_k<<<dim3(8192), blk, 0, stream>>>(bufA, S, C * S, gnm_w, gnm_b, stats);

  // 2) forward DHT with crop fused: z (96->24), y, x
  dht_axis_wmma<<<dim3(4608, 2), blk, 0, stream>>>(bufA, t1, Hf, 96, 24, 1);
  dht_axis_wmma<<<dim3(1152, 2), blk, 0, stream>>>(t1, t2, Hf, 96, 24, 24);
  dht_axis_wmma<<<dim3(288, 2), blk, 0, stream>>>(t2, xcA, Hf, 96, 24, 576);

  // 3) spectral block 0 (+GN+relu)
  spectral_block_k<<<dim3(P / 256, C), blk, 0, stream>>>(xcA, sw0, cw0, xcB, C, P);
  reset_stats_k<<<1, 1, 0, stream>>>(stats);
  gn_stats_k<<<dim3(432), blk, 0, stream>>>(xcB, C * P, stats);
  gn_apply_k<<<dim3(432), blk, 0, stream>>>(xcB, P, C * P, gn0_w, gn0_b, stats);

  // 4) spectral block 1 (+GN+relu)
  spectral_block_k<<<dim3(P / 256, C), blk, 0, stream>>>(xcB, sw1, cw1, xcA, C, P);
  reset_stats_k<<<1, 1, 0, stream>>>(stats);
  gn_stats_k<<<dim3(432), blk, 0, stream>>>(xcA, C * P, stats);
  gn_apply_k<<<dim3(432), blk, 0, stream>>>(xcA, P, C * P, gn1_w, gn1_b, stats);

  // 5) inverse DHT with pad fused (1/96 per axis in Hi): x (24->96), y, z
  dht_axis_wmma<<<dim3(288, 6), blk, 0, stream>>>(xcA, u1, Hi, 24, 96, 576);
  dht_axis_wmma<<<dim3(1152, 6), blk, 0, stream>>>(u1, u2, Hi, 24, 96, 24);
  dht_axis_wmma<<<dim3(4608, 6), blk, 0, stream>>>(u2, bufB, Hi, 24, 96, 1);

  reset_stats_k<<<1, 1, 0, stream>>>(stats);
  gn_stats_k<<<dim3(2048), blk, 0, stream>>>(bufB, C * S, stats);
  gn_apply_k<<<dim3(8192), blk, 0, stream>>>(bufB, S, C * S, gno_w, gno_b, stats);

  // 6) concat([bufB, bufA]) -> conv 128->64 + bias, then GN + relu (in place)
  conv1x1_wmma<<<dim3(6912, 4), blk, 0, stream>>>(bufB, bufA, Wc, bc, out, S, 2 * C, C, C);
  reset_stats_k<<<1, 1, 0, stream>>>(stats);
  gn_stats_k<<<dim3(2048), blk, 0, stream>>>(out, C * S, stats);
  gn_apply_k<<<dim3(8192), blk, 0, stream>>>(out, S, C * S, gnc_w, gnc_b, stats);
}